// GNNModel_3573412790951
// MI455X (gfx1250) — compile-verified
//
#include <hip/hip_runtime.h>
#include <hip/hip_bf16.h>

typedef __attribute__((ext_vector_type(2))) float v2f;
typedef __attribute__((ext_vector_type(8))) float v8f;

#define GCN_D 64

// ---------------- utility kernels ----------------

__global__ void zero_f32(float* p, int n) {
    int i = blockIdx.x * blockDim.x + threadIdx.x;
    if (i < n) p[i] = 0.0f;
}

// deg[dst[e]] += 1 for every real edge (self-loop contributes +1 in dinv kernel)
__global__ void deg_kernel(const int* __restrict__ dst, float* __restrict__ deg, int E) {
    int e = blockIdx.x * blockDim.x + threadIdx.x;
    if (e < E) atomicAdd(&deg[dst[e]], 1.0f);
}

// dinv[i] = rsqrt(deg_edges[i] + 1)   (self-loop included; always > 0)
__global__ void dinv_kernel(float* __restrict__ d, int n) {
    int i = blockIdx.x * blockDim.x + threadIdx.x;
    if (i < n) d[i] = rsqrtf(d[i] + 1.0f);
}

// ---------------- WMMA fp32 GEMM:  H[N,64] = X[N,64] @ W[64,64] ----------------
// One wave32 computes a 16x64 output strip using V_WMMA_F32_16X16X4_F32,
// K=64 as 16 chained k-steps of 4. EXEC is all-ones for every active wave
// (early-exit is wave-uniform).
__global__ void gemm_wmma_f32(const float* __restrict__ X,
                              const float* __restrict__ W,
                              float* __restrict__ H,
                              int nrows, int mtiles) {
    const int wave = blockIdx.x * (blockDim.x >> 5) + (threadIdx.x >> 5);
    if (wave >= mtiles) return;                    // uniform per wave
    const int lane = threadIdx.x & 31;
    const int l15  = lane & 15;
    const int hi   = lane >> 4;                    // 0: K+{0,1} half, 1: K+{2,3} half
    const int rowBase = wave * 16;
    int arow = rowBase + l15;
    if (arow >= nrows) arow = nrows - 1;           // clamp (values masked at store)

    v8f acc0 = {}, acc1 = {}, acc2 = {}, acc3 = {};

    const float* __restrict__ xrow = X + (size_t)arow * GCN_D;

    #pragma unroll
    for (int kk = 0; kk < GCN_D; kk += 4) {
        const int kb = kk + hi * 2;
        v2f a;
        a.x = xrow[kb];
        a.y = xrow[kb + 1];

        const float* __restrict__ wp = W + (size_t)kb * GCN_D + l15;
        v2f b0, b1, b2, b3;
        b0.x = wp[0];       b0.y = wp[GCN_D];
        b1.x = wp[16];      b1.y = wp[16 + GCN_D];
        b2.x = wp[32];      b2.y = wp[32 + GCN_D];
        b3.x = wp[48];      b3.y = wp[48 + GCN_D];

        acc0 = __builtin_amdgcn_wmma_f32_16x16x4_f32(false, a, false, b0, (short)0, acc0, false, false);
        acc1 = __builtin_amdgcn_wmma_f32_16x16x4_f32(false, a, false, b1, (short)0, acc1, false, false);
        acc2 = __builtin_amdgcn_wmma_f32_16x16x4_f32(false, a, false, b2, (short)0, acc2, false, false);
        acc3 = __builtin_amdgcn_wmma_f32_16x16x4_f32(false, a, false, b3, (short)0, acc3, false, false);
    }

    // C/D layout: VGPR r -> M = r (lanes 0-15) / r+8 (lanes 16-31), N = l15
    #pragma unroll
    for (int r = 0; r < 8; ++r) {
        const int row = rowBase + r + hi * 8;
        if (row < nrows) {
            float* __restrict__ o = H + (size_t)row * GCN_D + l15;
            o[0]  = acc0[r];
            o[16] = acc1[r];
            o[32] = acc2[r];
            o[48] = acc3[r];
        }
    }
}

// ---------------- layer-1 edge scatter: AGG[dst] += H[src] * dinv[s]*dinv[d] ----
// blockDim = (16, 16): threadIdx.x = feature-quad (float4), threadIdx.y = edge-in-block.
// One b128 gather + 4 f32 atomics per thread; 51.2M atomics is the algorithmic floor.
__global__ void scatter1_kernel(const int* __restrict__ src, const int* __restrict__ dst,
                                const float* __restrict__ H, const float* __restrict__ dinv,
                                float* __restrict__ AGG, int E) {
    int e = blockIdx.x * blockDim.y + threadIdx.y;
    if (e >= E) return;
    const int s = src[e];
    const int d = dst[e];
    const float nrm = dinv[s] * dinv[d];
    const int q = threadIdx.x;                     // 0..15 (quad of features)
    const float4 hv = ((const float4*)(H + (size_t)s * GCN_D))[q];
    float* __restrict__ base = AGG + (size_t)d * GCN_D + q * 4;
    atomicAdd(base + 0, hv.x * nrm);
    atomicAdd(base + 1, hv.y * nrm);
    atomicAdd(base + 2, hv.z * nrm);
    atomicAdd(base + 3, hv.w * nrm);
}

// Fused: layer-1 self-loop + bias + ReLU + dot with W2 (64x1).
// The layer-1 activations have exactly one consumer (h2), so never materialize them.
__global__ void finalize1_dot_kernel(const float* __restrict__ AGG, const float* __restrict__ H,
                                     const float* __restrict__ dinv, const float* __restrict__ b1,
                                     const float* __restrict__ W2, float* __restrict__ h2, int n) {
    int i = blockIdx.x * blockDim.x + threadIdx.x;
    if (i >= n) return;
    const float di = dinv[i];
    const float d2 = di * di;
    const float4* __restrict__ ag = (const float4*)(AGG + (size_t)i * GCN_D);
    const float4* __restrict__ hh = (const float4*)(H + (size_t)i * GCN_D);
    const float4* __restrict__ bb = (const float4*)b1;
    const float4* __restrict__ ww = (const float4*)W2;
    float s = 0.0f;
    #pragma unroll
    for (int q = 0; q < GCN_D / 4; ++q) {
        const float4 a = ag[q];
        const float4 h = hh[q];
        const float4 b = bb[q];
        const float4 w = ww[q];
        s += fmaxf(a.x + h.x * d2 + b.x, 0.0f) * w.x;
        s += fmaxf(a.y + h.y * d2 + b.y, 0.0f) * w.y;
        s += fmaxf(a.z + h.z * d2 + b.z, 0.0f) * w.z;
        s += fmaxf(a.w + h.w * d2 + b.w, 0.0f) * w.w;
    }
    h2[i] = s;
}

__global__ void scatter2_kernel(const int* __restrict__ src, const int* __restrict__ dst,
                                const float* __restrict__ h2, const float* __restrict__ dinv,
                                float* __restrict__ agg2, int E) {
    int e = blockIdx.x * blockDim.x + threadIdx.x;
    if (e >= E) return;
    const int s = src[e];
    const int d = dst[e];
    atomicAdd(&agg2[d], h2[s] * dinv[s] * dinv[d]);
}

__global__ void finalize2_kernel(const float* __restrict__ agg2, const float* __restrict__ h2,
                                 const float* __restrict__ dinv, const float* __restrict__ b2,
                                 float* __restrict__ out, int n) {
    int i = blockIdx.x * blockDim.x + threadIdx.x;
    if (i >= n) return;
    const float di = dinv[i];
    out[i] = agg2[i] + h2[i] * di * di + b2[0];
}

// ---------------- launch ----------------

extern "C" void kernel_launch(void* const* d_in, const int* in_sizes, int n_in,
                              void* d_out, int out_size, void* d_ws, size_t ws_size,
                              hipStream_t stream) {
    const float* x  = (const float*)d_in[0];
    const int*   ei = (const int*)d_in[1];     // [2, E] flat (JAX canonicalizes int64->int32)
    const float* W1 = (const float*)d_in[2];
    const float* b1 = (const float*)d_in[3];
    const float* W2 = (const float*)d_in[4];
    const float* b2 = (const float*)d_in[5];
    float* out = (float*)d_out;

    const int N = in_sizes[0] / GCN_D;         // 50000
    const int E = in_sizes[1] / 2;             // 800000
    const int* src = ei;
    const int* dst = ei + E;

    // workspace carve-out (~26 MB)
    float* H    = (float*)d_ws;                     // N*64
    float* AGG  = H    + (size_t)N * GCN_D;         // N*64
    float* dinv = AGG  + (size_t)N * GCN_D;         // N (deg, then dinv, in place)
    float* h2   = dinv + N;                         // N
    float* agg2 = h2   + N;                         // N

    const int T = 256;
    const int totNF = N * GCN_D;

    // 1. zero accumulators / degree
    zero_f32<<<(totNF + T - 1) / T, T, 0, stream>>>(AGG, totNF);
    zero_f32<<<(N + T - 1) / T, T, 0, stream>>>(dinv, N);
    zero_f32<<<(N + T - 1) / T, T, 0, stream>>>(agg2, N);

    // 2. degree + 3. dinv = rsqrt(deg+1)
    deg_kernel<<<(E + T - 1) / T, T, 0, stream>>>(dst, dinv, E);
    dinv_kernel<<<(N + T - 1) / T, T, 0, stream>>>(dinv, N);

    // 4. H = x @ W1  (fp32 WMMA, one wave per 16x64 strip)
    const int mtiles = (N + 15) / 16;               // 3125
    const int wavesPerBlk = T / 32;                 // 8
    const int gblocks = (mtiles + wavesPerBlk - 1) / wavesPerBlk;
    gemm_wmma_f32<<<gblocks, T, 0, stream>>>(x, W1, H, N, mtiles);

    // 5. edge scatter layer 1 (16 edges per 256-thread block, float4 gathers)
    {
        dim3 blk(16, 16);
        int grd = (E + 16 - 1) / 16;
        scatter1_kernel<<<grd, blk, 0, stream>>>(src, dst, H, dinv, AGG, E);
    }

    // 6+7. fused self-loop + bias + relu + dot(W2) -> h2
    finalize1_dot_kernel<<<(N + T - 1) / T, T, 0, stream>>>(AGG, H, dinv, b1, W2, h2, N);

    // 8. edge scatter layer 2 (scalar)
    scatter2_kernel<<<(E + T - 1) / T, T, 0, stream>>>(src, dst, h2, dinv, agg2, E);

    // 9. final self-loop + bias
    finalize2_kernel<<<(N + T - 1) / T, T, 0, stream>>>(agg2, h2, dinv, b2, out, N);
}